// ODE_RNN_42391327211900
// MI455X (gfx1250) — compile-verified
//
#include <hip/hip_runtime.h>
#include <hip/hip_bf16.h>
#include <math.h>

// ---------------------------------------------------------------------------
// ODE-RNN for MI455X (gfx1250, wave32).
//  Phase 1 (parallel):  U[H][T] = W_in @ X + b_in + b_h  via v_wmma_f32_16x16x4_f32
//  Phase 2 (sequential): persistent 32-WG kernel; W_f1/W_f2 row slices cached
//    in LDS via the Tensor Data Mover (tensor_load_to_lds + s_wait_tensorcnt),
//    h/t1 staged in LDS each phase, grid barrier between dependent matvecs.
// ---------------------------------------------------------------------------

typedef __attribute__((ext_vector_type(2))) float     v2f;
typedef __attribute__((ext_vector_type(8))) float     v8f;
typedef __attribute__((ext_vector_type(4))) unsigned  v4u;
typedef __attribute__((ext_vector_type(4))) int       v4i;
typedef __attribute__((ext_vector_type(8))) int       v8i;

#define T_STEPS 8192
#define IN_DIM  512
#define H_DIM   1024
#define OUT_DIM 256
#define NSUB    4

#define REC_WGS   32
#define REC_TPB   512                       // 16 wave32 waves per WG
#define REC_WAVES (REC_WGS * (REC_TPB / 32))   // 512 waves
#define ROWS_PER_WAVE (H_DIM / REC_WAVES)      // 2 rows per wave per matvec

// LDS: Wf1 slice [2*16 rows][1024] + Wf2 slice + vbuf[1024]
#define LDS_FLOATS (32 * H_DIM + 32 * H_DIM + H_DIM)
#define LDS_BYTES  (LDS_FLOATS * 4)

#if __has_builtin(__builtin_amdgcn_tensor_load_to_lds)
#define HAVE_TDM 1
#else
#define HAVE_TDM 0
#endif

// ---------------------------- small kernels --------------------------------

__global__ void transpose_x_kernel(const float* __restrict__ x, float* __restrict__ xT) {
    size_t idx = (size_t)blockIdx.x * blockDim.x + threadIdx.x;
    if (idx < (size_t)T_STEPS * IN_DIM) {
        int i = (int)(idx / IN_DIM);
        int k = (int)(idx % IN_DIM);
        xT[(size_t)k * T_STEPS + i] = x[idx];
    }
}

__global__ void transpose_win_kernel(const float* __restrict__ W, float* __restrict__ WT) {
    size_t idx = (size_t)blockIdx.x * blockDim.x + threadIdx.x;
    if (idx < (size_t)H_DIM * IN_DIM) {
        int m = (int)(idx / IN_DIM);
        int k = (int)(idx % IN_DIM);
        WT[(size_t)k * H_DIM + m] = W[idx];
    }
}

__global__ void init_kernel(float* __restrict__ h, unsigned* __restrict__ cnt) {
    int i = (int)(blockIdx.x * blockDim.x + threadIdx.x);
    if (i < H_DIM) h[i] = 0.0f;
    if (i == 0) { cnt[0] = 0u; cnt[1] = 0u; }   // barrier arrive count, generation
}

// ------------------- Phase 1: U = W_in @ X + b_in + b_h (WMMA fp32) --------
// Fragment layouts per CDNA5 ISA 7.12.2 (lane m|n = l&15, hi = l>>4):
//   A 16x4 f32 : VGPR v holds K = 2*hi + v        (coalesced via W_inT)
//   B 4x16 f32 : VGPR v holds K = 2*hi + v        (coalesced via xT)
//   C/D 16x16  : VGPR v holds M = v + 8*hi
__global__ __launch_bounds__(256)
void uin_gemm_kernel(const float* __restrict__ WinT,   // [IN][H]
                     const float* __restrict__ xT,     // [IN][T]
                     const float* __restrict__ b_in,
                     const float* __restrict__ b_h,
                     float* __restrict__ U)            // [H][T]
{
    const int wave = (int)(blockIdx.x * 8 + (threadIdx.x >> 5));
    const int lane = (int)(threadIdx.x & 31);
    const int m16  = lane & 15;
    const int hi   = lane >> 4;

    const int mt = wave & 63;      // 64 tiles over H
    const int nt = wave >> 6;      // 512 tiles over T
    const int mbase = mt * 16;
    const int nbase = nt * 16;

    v8f c = {};
    for (int k = 0; k < IN_DIM; k += 4) {
        const int ka = k + 2 * hi;
        v2f a, b;
        a.x = WinT[(size_t)ka       * H_DIM   + mbase + m16];
        a.y = WinT[(size_t)(ka + 1) * H_DIM   + mbase + m16];
        b.x = xT  [(size_t)ka       * T_STEPS + nbase + m16];
        b.y = xT  [(size_t)(ka + 1) * T_STEPS + nbase + m16];
        c = __builtin_amdgcn_wmma_f32_16x16x4_f32(false, a, false, b,
                                                  (short)0, c, false, false);
    }

#pragma unroll
    for (int v = 0; v < 8; ++v) {
        const int row = mbase + v + 8 * hi;
        U[(size_t)row * T_STEPS + nbase + m16] = c[v] + b_in[row] + b_h[row];
    }
}

// ------------------- Phase 2: persistent sequential recurrence -------------

// TDM: load a 16-row x 1024-col fp32 tile (rows contiguous, stride 1024) into LDS.
// D# per CDNA5 ISA 8.3/8.4: group0 = {count=1, lds_addr, global_addr, type=2},
// group1 = {data_size=4B, tensor_dim0=1024, tensor_dim1=1024, tile_dim0=1024,
//           tile_dim1=16, tensor_dim0_stride=1024}.
__device__ __forceinline__ void tdm_load_rows16(const float* gsrc, unsigned lds_off) {
#if HAVE_TDM
    unsigned long long ga = (unsigned long long)(uintptr_t)gsrc;
    v4u g0 = { 1u,                                   // count=1, user mode
               lds_off,                              // lds_addr (bytes)
               (unsigned)ga,                         // global_addr[31:0]
               ((unsigned)(ga >> 32) & 0x01FFFFFFu) | 0x80000000u }; // [56:32] | type=2
    v8i g1 = { (int)0x00020000,        // data_size=2 (4 bytes), no multicast
               (int)(1024u << 16),     // tensor_dim0 = 1024 (low 16 in [63:48])
               (int)(1024u << 16),     // tensor_dim0 hi=0, tensor_dim1 = 1024
               (int)(1024u << 16),     // tensor_dim1 hi=0, tile_dim0 = 1024
               16,                     // tile_dim1 = 16, tile_dim2 = 0
               1024,                   // tensor_dim0_stride = 1024
               0, 0 };                 // stride hi, tensor_dim1_stride = 0
    v4i z4 = { 0, 0, 0, 0 };
#if __clang_major__ >= 23
    v8i z8 = { 0, 0, 0, 0, 0, 0, 0, 0 };
    __builtin_amdgcn_tensor_load_to_lds(g0, g1, z4, z4, z8, 0);
#else
    __builtin_amdgcn_tensor_load_to_lds(g0, g1, z4, z4, 0);
#endif
#else
    (void)gsrc; (void)lds_off;
#endif
}

__device__ __forceinline__ float wave_dot4(const float* __restrict__ row,
                                           const float* __restrict__ v, int lane) {
    const float4* r4 = (const float4*)row;
    const float4* v4 = (const float4*)v;
    float s = 0.0f;
#pragma unroll
    for (int q = 0; q < H_DIM / 4 / 32; ++q) {       // 8 iterations, b128 loads
        float4 a = r4[q * 32 + lane];
        float4 b = v4[q * 32 + lane];
        s = fmaf(a.x, b.x, s); s = fmaf(a.y, b.y, s);
        s = fmaf(a.z, b.z, s); s = fmaf(a.w, b.w, s);
    }
#pragma unroll
    for (int off = 16; off > 0; off >>= 1)
        s += __shfl_xor(s, off, 32);
    return s;   // full sum in all lanes
}

__device__ __forceinline__ void stage_vec(float* __restrict__ dst,
                                          const float* __restrict__ src) {
    for (int k = threadIdx.x; k < H_DIM; k += REC_TPB) dst[k] = src[k];
    __syncthreads();
}

__device__ __forceinline__ void grid_sync(unsigned* __restrict__ cnt) {
    __threadfence();          // release this WG's h/t1 stores
    __syncthreads();
    if (threadIdx.x == 0) {
        volatile unsigned* gen = cnt + 1;
        unsigned g = *gen;
        if (atomicAdd(cnt, 1u) == (unsigned)(REC_WGS - 1)) {
            cnt[0] = 0u;
            __threadfence();
            *gen = g + 1u;
        } else {
            while (*gen == g) { __builtin_amdgcn_s_sleep(1); }
        }
    }
    __syncthreads();
    __threadfence();          // acquire other WGs' stores
}

__global__ __launch_bounds__(REC_TPB)
void rec_kernel(const float* __restrict__ t,
                const float* __restrict__ W_h,
                const float* __restrict__ W_h2, const float* __restrict__ b_h2,
                const float* __restrict__ W_f1, const float* __restrict__ b_f1,
                const float* __restrict__ W_f2, const float* __restrict__ b_f2,
                const float* __restrict__ W_dec, const float* __restrict__ b_dec,
                const float* __restrict__ U,     // [H][T], includes b_in+b_h
                float* __restrict__ h,
                float* __restrict__ t1,
                unsigned* __restrict__ cnt,
                float* __restrict__ out)
{
    extern __shared__ float lds[];
    float* Wf1c = lds;                    // [2 blocks][16 rows][1024]
    float* Wf2c = lds + 32 * H_DIM;
    float* vbuf = lds + 64 * H_DIM;       // staged h / t1 vector

    const int lane = (int)(threadIdx.x & 31);
    const int w    = (int)(threadIdx.x >> 5);              // wave-in-WG 0..15
    const int gw   = (int)(blockIdx.x * (REC_TPB / 32) + w);

    // ---- one-time preload of this WG's W_f1 / W_f2 row slices into LDS.
    // Block p holds global rows [p*512 + blockIdx.x*16, +16); LDS row = p*16+w.
#if HAVE_TDM
    if (threadIdx.x < 32) {               // wave 0 drives the TDM
        const unsigned lbase = (unsigned)(uintptr_t)lds;
        const size_t r0 = (size_t)blockIdx.x * 16;
        tdm_load_rows16(W_f1 + r0 * H_DIM,                 lbase);
        tdm_load_rows16(W_f1 + (r0 + 512) * H_DIM,         lbase + 16 * 4096);
        tdm_load_rows16(W_f2 + r0 * H_DIM,                 lbase + 32 * 4096);
        tdm_load_rows16(W_f2 + (r0 + 512) * H_DIM,         lbase + 48 * 4096);
#if __has_builtin(__builtin_amdgcn_s_wait_tensorcnt)
        __builtin_amdgcn_s_wait_tensorcnt(0);
#else
        asm volatile("s_wait_tensorcnt 0x0" ::: "memory");
#endif
    }
#else
    for (int lr = 0; lr < 32; ++lr) {
        int p = lr >> 4, rr = lr & 15;
        size_t r = (size_t)(p * 512 + blockIdx.x * 16 + rr);
        for (int k = threadIdx.x; k < H_DIM; k += REC_TPB) {
            Wf1c[(size_t)lr * H_DIM + k] = W_f1[r * H_DIM + k];
            Wf2c[(size_t)lr * H_DIM + k] = W_f2[r * H_DIM + k];
        }
    }
#endif
    __syncthreads();

    for (int i = 1; i < T_STEPS; ++i) {
        const float dt = (t[i] - t[i - 1]) * (1.0f / NSUB);

        // warm L0/L2 for this step's RNN-cell weight rows
#pragma unroll
        for (int p = 0; p < ROWS_PER_WAVE; ++p) {
            const size_t r = (size_t)(gw + p * REC_WAVES);
            __builtin_prefetch(W_h  + r * H_DIM + lane * 32, 0, 0);
            __builtin_prefetch(W_h2 + r * H_DIM + lane * 32, 0, 0);
        }

        // ---- 4 Euler substeps: h += dt*(W_f2 @ tanh(W_f1 @ h + b_f1) + b_f2)
        for (int s = 0; s < NSUB; ++s) {
            stage_vec(vbuf, h);
#pragma unroll
            for (int p = 0; p < ROWS_PER_WAVE; ++p) {
                const int r  = gw + p * REC_WAVES;
                const int lr = p * 16 + w;
                const float y = wave_dot4(Wf1c + (size_t)lr * H_DIM, vbuf, lane);
                if (lane == 0) t1[r] = tanhf(y + b_f1[r]);
            }
            grid_sync(cnt);

            stage_vec(vbuf, t1);
#pragma unroll
            for (int p = 0; p < ROWS_PER_WAVE; ++p) {
                const int r  = gw + p * REC_WAVES;
                const int lr = p * 16 + w;
                const float y = wave_dot4(Wf2c + (size_t)lr * H_DIM, vbuf, lane);
                if (lane == 0) h[r] = fmaf(dt, y + b_f2[r], h[r]);
            }
            grid_sync(cnt);
        }

        // ---- RNN cell: h = tanh(W_h2 @ tanh(U[:,i] + W_h @ h) + b_h2)
        stage_vec(vbuf, h);
#pragma unroll
        for (int p = 0; p < ROWS_PER_WAVE; ++p) {
            const int r = gw + p * REC_WAVES;
            const float y = wave_dot4(W_h + (size_t)r * H_DIM, vbuf, lane);
            if (lane == 0) t1[r] = tanhf(U[(size_t)r * T_STEPS + i] + y);
        }
        grid_sync(cnt);

        stage_vec(vbuf, t1);
#pragma unroll
        for (int p = 0; p < ROWS_PER_WAVE; ++p) {
            const int r = gw + p * REC_WAVES;
            const float y = wave_dot4(W_h2 + (size_t)r * H_DIM, vbuf, lane);
            if (lane == 0) h[r] = tanhf(y + b_h2[r]);
        }
        grid_sync(cnt);
    }

    // ---- decoder: out = W_dec @ h_T + b_dec (256 rows, one per wave)
    stage_vec(vbuf, h);
    if (gw < OUT_DIM) {
        const float y = wave_dot4(W_dec + (size_t)gw * H_DIM, vbuf, lane);
        if (lane == 0) out[gw] = y + b_dec[gw];
    }
}

// ---------------------------- host glue ------------------------------------

extern "C" void kernel_launch(void* const* d_in, const int* in_sizes, int n_in,
                              void* d_out, int out_size, void* d_ws, size_t ws_size,
                              hipStream_t stream) {
    const float* t     = (const float*)d_in[0];
    const float* x     = (const float*)d_in[1];
    const float* W_in  = (const float*)d_in[2];
    const float* b_in  = (const float*)d_in[3];
    const float* W_h   = (const float*)d_in[4];
    const float* b_h   = (const float*)d_in[5];
    const float* W_h2  = (const float*)d_in[6];
    const float* b_h2  = (const float*)d_in[7];
    const float* W_f1  = (const float*)d_in[8];
    const float* b_f1  = (const float*)d_in[9];
    const float* W_f2  = (const float*)d_in[10];
    const float* b_f2  = (const float*)d_in[11];
    const float* W_dec = (const float*)d_in[12];
    const float* b_dec = (const float*)d_in[13];
    float* out = (float*)d_out;

    float* ws   = (float*)d_ws;
    float* U    = ws;                                    // H*T      = 8M  f32
    float* xT   = U    + (size_t)H_DIM * T_STEPS;        // IN*T     = 4M  f32
    float* WinT = xT   + (size_t)IN_DIM * T_STEPS;       // IN*H     = 512K f32
    float* hbuf = WinT + (size_t)IN_DIM * H_DIM;         // H
    float* tbuf = hbuf + H_DIM;                          // H
    unsigned* cnt = (unsigned*)(tbuf + H_DIM);           // [arrive, generation]

    // allow 260KB dynamic LDS (CDNA5: up to 320KB per workgroup)
    (void)hipFuncSetAttribute((const void*)rec_kernel,
                              hipFuncAttributeMaxDynamicSharedMemorySize, LDS_BYTES);

    // 1) transposes for coalesced WMMA fragment loads
    {
        size_t n = (size_t)T_STEPS * IN_DIM;
        transpose_x_kernel<<<(unsigned)((n + 255) / 256), 256, 0, stream>>>(x, xT);
        size_t m = (size_t)H_DIM * IN_DIM;
        transpose_win_kernel<<<(unsigned)((m + 255) / 256), 256, 0, stream>>>(W_in, WinT);
    }

    // 2) state / barrier init (runs every launch -> deterministic)
    init_kernel<<<4, 256, 0, stream>>>(hbuf, cnt);

    // 3) WMMA GEMM: U = W_in @ X + b_in + b_h (64 Mtiles x 512 Ntiles, 1 wave/tile)
    {
        const int tiles = 64 * 512;           // 32768 waves
        uin_gemm_kernel<<<tiles / 8, 256, 0, stream>>>(WinT, xT, b_in, b_h, U);
    }

    // 4) persistent sequential recurrence + decoder
    rec_kernel<<<REC_WGS, REC_TPB, LDS_BYTES, stream>>>(t, W_h, W_h2, b_h2,
                                                        W_f1, b_f1, W_f2, b_f2,
                                                        W_dec, b_dec,
                                                        U, hbuf, tbuf, cnt, out);
}